// PAM_Module_11974368821924
// MI455X (gfx1250) — compile-verified
//
#include <hip/hip_runtime.h>

// PAM axial attention for MI455X (gfx1250, wave32).
// f32 end-to-end; GEMMs via V_WMMA_F32_16X16X4_F32; attention-matrix tiles
// staged into LDS with the Tensor Data Mover (tensor_load_to_lds + TENSORcnt).

#define B_  8
#define C_  64
#define CQ_ 8

typedef __attribute__((ext_vector_type(2))) float    v2f;
typedef __attribute__((ext_vector_type(8))) float    v8f;
typedef __attribute__((ext_vector_type(4))) unsigned v4u;
typedef __attribute__((ext_vector_type(8))) unsigned v8u;

__device__ __forceinline__ v8f wmma4(v2f a, v2f b, v8f c) {
  // (neg_a, A, neg_b, B, c_mod, C, reuse_a, reuse_b)
  return __builtin_amdgcn_wmma_f32_16x16x4_f32(false, a, false, b, (short)0, c,
                                               false, false);
}

__device__ __forceinline__ v8f vzero8() {
  v8f z;
#pragma unroll
  for (int i = 0; i < 8; ++i) z[i] = 0.0f;
  return z;
}

// ---------------------------------------------------------------------------
// TDM: DMA a 2D tile (tile1 rows x tile0 elems, row stride stride0 elems, f32)
// from global memory into LDS at ldsOff, inserting padAmt dwords of padding
// every (1<<(padIvlCode+1)) dwords so LDS rows land bank-conflict free.
// 2-operand form => descriptor groups 2/3 NULL (2D tensor). Issue from one
// wave; completion tracked with TENSORcnt.
// ---------------------------------------------------------------------------
__device__ __forceinline__ void tdm_load_2d(const float* gaddr, unsigned ldsOff,
                                            unsigned tile0, unsigned tile1,
                                            unsigned stride0,
                                            unsigned padIvlCode,
                                            unsigned padAmtCode) {
  unsigned long long ga = (unsigned long long)(uintptr_t)gaddr;
  v4u g0;
  g0[0] = 1u;                                   // count=1, user descriptor
  g0[1] = ldsOff;                               // lds_addr (bytes)
  g0[2] = (unsigned)(ga & 0xffffffffu);         // global_addr[31:0]
  g0[3] = (unsigned)((ga >> 32) & 0x1ffffffu)   // global_addr[56:32]
          | (2u << 30);                         // type = 2 ("image")
  v8u g1;
  g1[0] = (2u << 16)                            // data_size = 4B
          | (1u << 20)                          // pad_enable
          | (padIvlCode << 22)                  // pad_interval
          | (padAmtCode << 25);                 // pad_amount
  g1[1] = (tile0 & 0xffffu) << 16;              // tensor_dim0 = tile0 (tight)
  g1[2] = (tile0 >> 16) | ((tile1 & 0xffffu) << 16);  // tensor_dim1 = tile1
  g1[3] = (tile1 >> 16) | (tile0 << 16);        // tile_dim0
  g1[4] = tile1 & 0xffffu;                      // tile_dim1 (tile_dim2 = 0)
  g1[5] = stride0;                              // tensor_dim0_stride[31:0]
  g1[6] = 0u;                                   // stride0[47:32], stride1 lo
  g1[7] = 0u;
  asm volatile("tensor_load_to_lds %0, %1" :: "s"(g0), "s"(g1) : "memory");
}

// ---------------------------------------------------------------------------
// 1x1 conv producing q (Cq), k (Cq), v (C) per pixel. One pass over x.
// ---------------------------------------------------------------------------
__global__ __launch_bounds__(256) void conv_qkv_kernel(
    const float* __restrict__ x,
    const float* __restrict__ wq, const float* __restrict__ bq,
    const float* __restrict__ wk, const float* __restrict__ bk,
    const float* __restrict__ wv, const float* __restrict__ bv,
    float* __restrict__ q, float* __restrict__ k, float* __restrict__ v,
    int HW, int total)
{
  int idx = blockIdx.x * blockDim.x + threadIdx.x;
  if (idx >= total) return;
  int b   = idx / HW;
  int pix = idx - b * HW;
  const float* xb = x + (size_t)b * C_ * HW + pix;

  float aq[CQ_], ak[CQ_], av[C_];
#pragma unroll
  for (int o = 0; o < CQ_; ++o) { aq[o] = bq[o]; ak[o] = bk[o]; }
#pragma unroll
  for (int o = 0; o < C_; ++o) av[o] = bv[o];

  for (int c = 0; c < C_; ++c) {
    float xv = xb[(size_t)c * HW];
#pragma unroll
    for (int o = 0; o < CQ_; ++o) {
      aq[o] = fmaf(wq[o * C_ + c], xv, aq[o]);
      ak[o] = fmaf(wk[o * C_ + c], xv, ak[o]);
    }
#pragma unroll
    for (int o = 0; o < C_; ++o) av[o] = fmaf(wv[o * C_ + c], xv, av[o]);
  }
#pragma unroll
  for (int o = 0; o < CQ_; ++o) {
    q[((size_t)b * CQ_ + o) * HW + pix] = aq[o];
    k[((size_t)b * CQ_ + o) * HW + pix] = ak[o];
  }
#pragma unroll
  for (int o = 0; o < C_; ++o)
    v[((size_t)b * C_ + o) * HW + pix] = av[o];
}

// ---------------------------------------------------------------------------
// Attention logits: L[b,p,p'] = sum_d Q[off(d,p)] * K[off(d,p')]
// off(d,p) = (d>>logN)*HW + (d & (N-1))*sInner + p*sP
//   width:  sInner=N, sP=1 (d=(cq,h), p=w)   height: sInner=1, sP=N
// One wave computes a 32x32 tile (2x2 WMMA register blocking).
// ---------------------------------------------------------------------------
__global__ __launch_bounds__(256) void logits_kernel(
    const float* __restrict__ Q, const float* __restrict__ K,
    float* __restrict__ L,
    int N, int logN, int HW, int sInner, int sP, int nT, int Dtot)
{
  int wave = (blockIdx.x * blockDim.x + threadIdx.x) >> 5;
  int lane = threadIdx.x & 31;
  int tilesPerB = nT * nT;
  if (wave >= B_ * tilesPerB) return;
  int b  = wave / tilesPerB;
  int t  = wave - b * tilesPerB;
  int tm = t / nT, tn = t - tm * nT;
  int q0 = tm * 32, k0 = tn * 32;
  int lm = lane & 15, hb = lane >> 4;

  const float* Qb = Q + (size_t)b * CQ_ * HW;
  const float* Kb = K + (size_t)b * CQ_ * HW;

  v8f acc00 = vzero8(), acc01 = vzero8(), acc10 = vzero8(), acc11 = vzero8();
  int mask = N - 1;
  int pa0 = (q0 + lm) * sP, pa1 = (q0 + 16 + lm) * sP;
  int pb0 = (k0 + lm) * sP, pb1 = (k0 + 16 + lm) * sP;

  for (int d0 = 0; d0 < Dtot; d0 += 4) {
    int dA   = d0 + 2 * hb;
    int off0 = (dA >> logN) * HW + (dA & mask) * sInner;
    int off1 = off0 + sInner;                 // dA+1 never crosses a cq block
    v2f a0 = { Qb[off0 + pa0], Qb[off1 + pa0] };
    v2f a1 = { Qb[off0 + pa1], Qb[off1 + pa1] };
    v2f b0 = { Kb[off0 + pb0], Kb[off1 + pb0] };
    v2f b1 = { Kb[off0 + pb1], Kb[off1 + pb1] };
    acc00 = wmma4(a0, b0, acc00);
    acc01 = wmma4(a0, b1, acc01);
    acc10 = wmma4(a1, b0, acc10);
    acc11 = wmma4(a1, b1, acc11);
  }

  float* Lb = L + (size_t)b * N * N;
#pragma unroll
  for (int r = 0; r < 8; ++r) {
    int row0 = q0 + r + 8 * hb;
    int row1 = row0 + 16;
    Lb[(size_t)row0 * N + k0 + lm]      = acc00[r];
    Lb[(size_t)row0 * N + k0 + 16 + lm] = acc01[r];
    Lb[(size_t)row1 * N + k0 + lm]      = acc10[r];
    Lb[(size_t)row1 * N + k0 + 16 + lm] = acc11[r];
  }
}

// ---------------------------------------------------------------------------
// Row softmax, wave32 per row; output layout A[k*osK + q*osQ].
// ---------------------------------------------------------------------------
__global__ __launch_bounds__(256) void softmax_kernel(
    const float* __restrict__ L, float* __restrict__ A,
    int N, int osK, int osQ, int rows)
{
  int wave = (blockIdx.x * blockDim.x + threadIdx.x) >> 5;
  int lane = threadIdx.x & 31;
  if (wave >= rows) return;
  int b = wave / N, qr = wave - b * N;
  const float* Lr = L + ((size_t)b * N + qr) * N;
  int cnt = N >> 5;

  float vals[8];
  float m = -3.0e38f;
#pragma unroll
  for (int j = 0; j < 8; ++j) {
    vals[j] = (j < cnt) ? Lr[lane + j * 32] : -3.0e38f;
    m = fmaxf(m, vals[j]);
  }
#pragma unroll
  for (int s = 16; s > 0; s >>= 1) m = fmaxf(m, __shfl_xor(m, s, 32));

  float sum = 0.0f;
#pragma unroll
  for (int j = 0; j < 8; ++j) { vals[j] = __expf(vals[j] - m); sum += vals[j]; }
#pragma unroll
  for (int s = 16; s > 0; s >>= 1) sum += __shfl_xor(sum, s, 32);
  float inv = 1.0f / sum;

  float* Ab = A + (size_t)b * N * N;
#pragma unroll
  for (int j = 0; j < 8; ++j)
    if (j < cnt) Ab[(size_t)(lane + j * 32) * osK + (size_t)qr * osQ] = vals[j] * inv;
}

// ---------------------------------------------------------------------------
// Width apply: out[b,(c,h),q] = 2*x + scale * sum_k V[(c,h),k] * At[k,q]
// Block = 8 waves sharing one 32-column slice of At (TDM-staged in LDS,
// padded rows stride 48 dwords -> conflict-free B-fragment reads).
// Wave w handles M-rows (mg*8+w)*32 .. +31; V rows are k-contiguous (v2f).
// ---------------------------------------------------------------------------
#define WPAD_ 48   // LDS row stride in dwords for the 32-wide At tile
__global__ __launch_bounds__(256) void apply_w_kernel(
    const float* __restrict__ V, const float* __restrict__ At,
    const float* __restrict__ x, float* __restrict__ out,
    const float* __restrict__ sc,
    int N, int HW, int nTm, int nTn)
{
  extern __shared__ float sh[];
  int nMG  = nTm >> 3;
  int perB = nMG * nTn;
  int blk  = blockIdx.x;
  int b  = blk / perB;
  int r  = blk - b * perB;
  int mg = r / nTn, tn = r - mg * nTn;
  int wv   = threadIdx.x >> 5;
  int lane = threadIdx.x & 31;
  int m0 = (mg * 8 + wv) * 32, n0 = tn * 32;
  int lm = lane & 15, hb = lane >> 4;

  const float* Vb = V + (size_t)b * C_ * HW;
  const float* Ab = At + (size_t)b * HW;
  const float* xb = x + (size_t)b * C_ * HW;
  float*       ob = out + (size_t)b * C_ * HW;
  float scale = sc[0];

  // Stage At[:, n0..n0+31] (N rows x 32 cols, row stride N) -> LDS.
  // pad_interval = 32 dwords (code 4), pad_amount = 16 dwords (code 15).
  if (threadIdx.x < 32) {
    tdm_load_2d(Ab + n0, (unsigned)(uintptr_t)sh, 32u, (unsigned)N,
                (unsigned)N, 4u, 15u);
    __builtin_amdgcn_s_wait_tensorcnt(0);
  }
  __syncthreads();

  v8f acc00 = vzero8(), acc01 = vzero8(), acc10 = vzero8(), acc11 = vzero8();
  for (int k0 = 0; k0 < N; k0 += 4) {
    int kk = k0 + 2 * hb;
    v2f a0 = *(const v2f*)(Vb + (size_t)(m0 + lm) * N + kk);
    v2f a1 = *(const v2f*)(Vb + (size_t)(m0 + 16 + lm) * N + kk);
    const float* Sr = sh + kk * WPAD_;
    v2f b0 = { Sr[lm],      Sr[WPAD_ + lm] };
    v2f b1 = { Sr[16 + lm], Sr[WPAD_ + 16 + lm] };
    acc00 = wmma4(a0, b0, acc00);
    acc01 = wmma4(a0, b1, acc01);
    acc10 = wmma4(a1, b0, acc10);
    acc11 = wmma4(a1, b1, acc11);
  }
#pragma unroll
  for (int r8 = 0; r8 < 8; ++r8) {
    int row0 = m0 + r8 + 8 * hb;
    int row1 = row0 + 16;
    size_t o00 = (size_t)row0 * N + n0 + lm;
    size_t o01 = o00 + 16;
    size_t o10 = (size_t)row1 * N + n0 + lm;
    size_t o11 = o10 + 16;
    ob[o00] = 2.0f * xb[o00] + scale * acc00[r8];
    ob[o01] = 2.0f * xb[o01] + scale * acc01[r8];
    ob[o10] = 2.0f * xb[o10] + scale * acc10[r8];
    ob[o11] = 2.0f * xb[o11] + scale * acc11[r8];
  }
}

// ---------------------------------------------------------------------------
// Height apply: out[b,c,h,w] += scale * sum_k Ah[h,k] * V[b,c,k,w]
// Block = 8 waves = 8 consecutive channels sharing one 32-row Ah slice
// (TDM-staged, padded rows stride N+4 dwords -> both lane halves of the
// b64 A-fragment reads spread across all 64 banks).
// ---------------------------------------------------------------------------
__global__ __launch_bounds__(256) void apply_h_kernel(
    const float* __restrict__ V, const float* __restrict__ Ah,
    float* __restrict__ out, const float* __restrict__ sc,
    int N, int logN, int HW, int nT)
{
  extern __shared__ float sh[];
  int tiles = nT * nT;
  int perB  = 8 * tiles;                 // 8 channel-groups x tiles
  int blk = blockIdx.x;
  int b  = blk / perB;
  int r  = blk - b * perB;
  int cg = r / tiles;
  int t  = r - cg * tiles;
  int tm = t / nT, tn = t - tm * nT;
  int wv   = threadIdx.x >> 5;
  int lane = threadIdx.x & 31;
  int c  = cg * 8 + wv;
  int m0 = tm * 32, n0 = tn * 32;
  int lm = lane & 15, hb = lane >> 4;
  int pitch = N + 4;

  const float* Vbc = V + ((size_t)b * C_ + c) * HW;
  const float* Ab  = Ah + (size_t)b * HW;
  float*       ob  = out + ((size_t)b * C_ + c) * HW;
  float scale = sc[0];

  // Stage Ah[m0..m0+31][:] (32 rows x N cols, contiguous) -> LDS.
  // pad_interval = N dwords (code logN-1), pad_amount = 4 dwords (code 3).
  if (threadIdx.x < 32) {
    tdm_load_2d(Ab + (size_t)m0 * N, (unsigned)(uintptr_t)sh, (unsigned)N, 32u,
                (unsigned)N, (unsigned)(logN - 1), 3u);
    __builtin_amdgcn_s_wait_tensorcnt(0);
  }
  __syncthreads();

  v8f acc00 = vzero8(), acc01 = vzero8(), acc10 = vzero8(), acc11 = vzero8();
  for (int k0 = 0; k0 < N; k0 += 4) {
    int kk = k0 + 2 * hb;
    v2f a0 = *(const v2f*)(sh + lm * pitch + kk);
    v2f a1 = *(const v2f*)(sh + (16 + lm) * pitch + kk);
    const float* Vr = Vbc + (size_t)kk * N;
    v2f b0 = { Vr[n0 + lm],      Vr[N + n0 + lm] };
    v2f b1 = { Vr[n0 + 16 + lm], Vr[N + n0 + 16 + lm] };
    acc00 = wmma4(a0, b0, acc00);
    acc01 = wmma4(a0, b1, acc01);
    acc10 = wmma4(a1, b0, acc10);
    acc11 = wmma4(a1, b1, acc11);
  }
#pragma unroll
  for (int r8 = 0; r8 < 8; ++r8) {
    int row0 = m0 + r8 + 8 * hb;
    int row1 = row0 + 16;
    size_t o00 = (size_t)row0 * N + n0 + lm;
    size_t o01 = o00 + 16;
    size_t o10 = (size_t)row1 * N + n0 + lm;
    size_t o11 = o10 + 16;
    ob[o00] += scale * acc00[r8];
    ob[o01] += scale * acc01[r8];
    ob[o10] += scale * acc10[r8];
    ob[o11] += scale * acc11[r8];
  }
}

// ---------------------------------------------------------------------------
extern "C" void kernel_launch(void* const* d_in, const int* in_sizes, int n_in,
                              void* d_out, int out_size, void* d_ws, size_t ws_size,
                              hipStream_t stream) {
  (void)in_sizes; (void)n_in; (void)out_size; (void)ws_size;
  const float* tmap = (const float*)d_in[0];
  const float* smap = (const float*)d_in[1];
  const float* tqw = (const float*)d_in[2];  const float* tqb = (const float*)d_in[3];
  const float* tkw = (const float*)d_in[4];  const float* tkb = (const float*)d_in[5];
  const float* tvw = (const float*)d_in[6];  const float* tvb = (const float*)d_in[7];
  const float* sqw = (const float*)d_in[8];  const float* sqb = (const float*)d_in[9];
  const float* skw = (const float*)d_in[10]; const float* skb = (const float*)d_in[11];
  const float* svw = (const float*)d_in[12]; const float* svb = (const float*)d_in[13];
  const float* alpha = (const float*)d_in[14];
  const float* beta  = (const float*)d_in[15];
  const float* gamma = (const float*)d_in[16];
  const float* omega = (const float*)d_in[17];

  const int NT = 64,  HWT = NT * NT;    // template H=W=64
  const int NS = 256, HWS = NS * NS;    // scene    H=W=256

  float* outT = (float*)d_out;
  float* outS = outT + (size_t)B_ * C_ * HWT;

  float* w = (float*)d_ws;
  size_t off = 0;
  float* qT  = w + off; off += (size_t)B_ * CQ_ * HWT;
  float* kT  = w + off; off += (size_t)B_ * CQ_ * HWT;
  float* vT  = w + off; off += (size_t)B_ * C_  * HWT;
  float* qS  = w + off; off += (size_t)B_ * CQ_ * HWS;
  float* kS  = w + off; off += (size_t)B_ * CQ_ * HWS;
  float* vS  = w + off; off += (size_t)B_ * C_  * HWS;
  float* LwT = w + off; off += (size_t)B_ * HWT;
  float* AwT = w + off; off += (size_t)B_ * HWT;
  float* LhT = w + off; off += (size_t)B_ * HWT;
  float* AhT = w + off; off += (size_t)B_ * HWT;
  float* LwS = w + off; off += (size_t)B_ * HWS;
  float* AwS = w + off; off += (size_t)B_ * HWS;
  float* LhS = w + off; off += (size_t)B_ * HWS;
  float* AhS = w + off; off += (size_t)B_ * HWS;

  dim3 blk(256);

  // 1) convs (one pass over each map)
  conv_qkv_kernel<<<(B_ * HWT) / 256, blk, 0, stream>>>(
      tmap, tqw, tqb, tkw, tkb, tvw, tvb, qT, kT, vT, HWT, B_ * HWT);
  conv_qkv_kernel<<<(B_ * HWS) / 256, blk, 0, stream>>>(
      smap, sqw, sqb, skw, skb, svw, svb, qS, kS, vS, HWS, B_ * HWS);

  // 2) logits (WMMA)
  logits_kernel<<<(8 * 2 * 2) / 8, blk, 0, stream>>>(qT, kT, LwT, NT, 6, HWT, NT, 1, 2, CQ_ * NT);
  logits_kernel<<<(8 * 2 * 2) / 8, blk, 0, stream>>>(qT, kT, LhT, NT, 6, HWT, 1, NT, 2, CQ_ * NT);
  logits_kernel<<<(8 * 8 * 8) / 8, blk, 0, stream>>>(qS, kS, LwS, NS, 8, HWS, NS, 1, 8, CQ_ * NS);
  logits_kernel<<<(8 * 8 * 8) / 8, blk, 0, stream>>>(qS, kS, LhS, NS, 8, HWS, 1, NS, 8, CQ_ * NS);

  // 3) softmax: width branch transposed (A^T[k][q]), height natural.
  softmax_kernel<<<(B_ * NT) / 8, blk, 0, stream>>>(LwT, AwT, NT, NT, 1, B_ * NT);
  softmax_kernel<<<(B_ * NT) / 8, blk, 0, stream>>>(LhT, AhT, NT, 1, NT, B_ * NT);
  softmax_kernel<<<(B_ * NS) / 8, blk, 0, stream>>>(LwS, AwS, NS, NS, 1, B_ * NS);
  softmax_kernel<<<(B_ * NS) / 8, blk, 0, stream>>>(LhS, AhS, NS, 1, NS, B_ * NS);

  // 4) width apply (TDM-staged At slice; out = 2x + scale*V A^T)
  apply_w_kernel<<<8 * ((C_ * NT / 32) / 8) * (NT / 32), blk,
                   (size_t)WPAD_ * NT * 4, stream>>>(
      vT, AwT, tmap, outT, alpha, NT, HWT, C_ * NT / 32, NT / 32);
  apply_w_kernel<<<8 * ((C_ * NS / 32) / 8) * (NS / 32), blk,
                   (size_t)WPAD_ * NS * 4, stream>>>(
      vS, AwS, smap, outS, beta, NS, HWS, C_ * NS / 32, NS / 32);

  // 5) height apply (TDM-staged Ah slice; out += scale * Ah V)
  apply_h_kernel<<<8 * 8 * (NT / 32) * (NT / 32), blk,
                   (size_t)(NT + 4) * 32 * 4, stream>>>(
      vT, AhT, outT, gamma, NT, 6, HWT, NT / 32);
  apply_h_kernel<<<8 * 8 * (NS / 32) * (NS / 32), blk,
                   (size_t)(NS + 4) * 32 * 4, stream>>>(
      vS, AhS, outS, omega, NS, 8, HWS, NS / 32);
}